// LCALSTM_59949153518079
// MI455X (gfx1250) — compile-verified
//
#include <hip/hip_runtime.h>
#include <hip/hip_bf16.h>

// ---------------- problem constants ----------------
#define H      1024
#define DLEN   65536
#define IN_LEN 1025
#define G4     4098          // 4H + 2
#define CMPT   0.8f
#define DT_F   0.6f
#define NCYC   10
#define EPSF   1e-8f

// ---------------- output layout (floats) ----------------
#define OUT_PI   0
#define OUT_VAL  4
#define OUT_H2   5
#define OUT_CM   (5 + H)
#define OUT_MT   (5 + 2 * H)
#define OUT_MEM  (5 + 3 * H)

// ---------------- workspace layout (floats) ----------------
#define WS_PREACT 0          // 4098
#define WS_CT     4224       // 1024  (q)
#define WS_HT     5248       // 1024
#define WS_DEC    6272       // 1024
#define WS_SCAL   7296       // [0]=inps_t  [1]=||q||
#define WS_SIMS   8192       // 65536
#define WS_WTS    73728      // 65536
#define WS_MTPART 139264     // 256*1024 partial m_t
#define WS_DEC2   401408     // 1024
#define WS_HT2    402432     // 1024

typedef float    v2f  __attribute__((ext_vector_type(2)));
typedef float    v4f  __attribute__((ext_vector_type(4)));
typedef float    v8f  __attribute__((ext_vector_type(8)));

#if defined(__AMDGCN__) && __has_builtin(__builtin_amdgcn_wmma_f32_16x16x4_f32)
#define USE_WMMA_F32X4 1
#endif

__device__ __forceinline__ float sigm(float x) { return 1.0f / (1.0f + expf(-x)); }

__device__ __forceinline__ float waveRed(float x) {
#pragma unroll
  for (int o = 16; o > 0; o >>= 1) x += __shfl_xor(x, o, 32);
  return x;
}

__device__ __forceinline__ float blockRed(float x, float* lds32) {
  const int lane = threadIdx.x & 31, w = threadIdx.x >> 5;
  x = waveRed(x);
  if (lane == 0) lds32[w] = x;
  __syncthreads();
  const int nw = (blockDim.x + 31) >> 5;
  float t = (threadIdx.x < (unsigned)nw) ? lds32[threadIdx.x] : 0.0f;
  if (w == 0) { t = waveRed(t); if (lane == 0) lds32[0] = t; }
  __syncthreads();
  const float r = lds32[0];
  __syncthreads();
  return r;
}

// ---- K1: preact = Wi2h@x + bi2h + Wh2h@h + bh2h  (wave per row) ----
__global__ void k_preact(const float* __restrict__ Wi2h, const float* __restrict__ bi2h,
                         const float* __restrict__ Wh2h, const float* __restrict__ bh2h,
                         const float* __restrict__ x, const float* __restrict__ h,
                         float* __restrict__ pre) {
  const int lane = threadIdx.x & 31, w = threadIdx.x >> 5;
  const int row = blockIdx.x * 8 + w;
  if (row >= G4) return;
  const float* w1 = Wi2h + (size_t)row * IN_LEN;
  const float* w2 = Wh2h + (size_t)row * H;
  float acc = 0.0f;
  for (int j = lane; j < IN_LEN; j += 32) acc += w1[j] * x[j];
  for (int j = lane; j < H; j += 32)      acc += w2[j] * h[j];
  acc = waveRed(acc);
  if (lane == 0) pre[row] = acc + bi2h[row] + bh2h[row];
}

// ---- K2: gates, c_t, h_t, ||q|| ----
__global__ __launch_bounds__(1024) void k_gates(const float* __restrict__ pre,
                                                const float* __restrict__ cprev,
                                                float* __restrict__ c_t,
                                                float* __restrict__ h_t,
                                                float* __restrict__ scal) {
  __shared__ float red[32];
  const int i = threadIdx.x;
  const float f  = sigm(pre[i]);
  const float o  = sigm(pre[H + i]);
  const float ig = sigm(pre[2 * H + i]);
  const float cn = tanhf(pre[3 * H + 2 + i]);
  const float c = cprev[i] * f + ig * cn;
  c_t[i] = c;
  h_t[i] = o * tanhf(c);
  const float ss = blockRed(c * c, red);
  if (i == 0) scal[1] = sqrtf(ss);
}

// ---- generic relu(W@x + b), wave per row (Wih is 1024x1024) ----
__global__ void k_matvec_relu(const float* __restrict__ W, const float* __restrict__ b,
                              const float* __restrict__ x, float* __restrict__ y,
                              int rows, int cols) {
  const int lane = threadIdx.x & 31, w = threadIdx.x >> 5;
  const int row = blockIdx.x * 8 + w;
  if (row >= rows) return;
  const float* wr = W + (size_t)row * cols;
  float acc = 0.0f;
  for (int j = lane; j < cols; j += 32) acc += wr[j] * x[j];
  acc = waveRed(acc);
  if (lane == 0) y[row] = fmaxf(acc + b[row], 0.0f);
}

// ---- K4: inps_t = sigmoid(Whpc[0] . [c_t ; dec_act] + bhpc[0]) ----
__global__ void k_phi(const float* __restrict__ c_t, const float* __restrict__ dec,
                      const float* __restrict__ Whpc, const float* __restrict__ bhpc,
                      float* __restrict__ scal) {
  __shared__ float red[32];
  const int t = threadIdx.x;
  float p = 0.0f;
  for (int j = t; j < 2 * H; j += 256) {
    const float v = (j < H) ? c_t[j] : dec[j - H];
    p += Whpc[j] * v;
  }
  p = blockRed(p, red);
  if (t == 0) scal[0] = sigm(p + bhpc[0]);
}

// ---- K5: sims = (mem @ q) / (||mem_row||*||q|| + eps)  via f32 WMMA -----
// One wave owns 16 rows of mem (A tile 16x4 f32: lanes 0-15 supply K=0,1 of row m,
// lanes 16-31 supply K=2,3 of row m). Each lane loads a 16B float4 spanning 8 K
// values; a single shfl_xor(16) exchange provides the cross-half sub-pairs, feeding
// TWO wmmas per load. B carries q in column 0 only, built branchlessly from
// per-lane selector constants (no EXEC-mask churn in the hot loop).
__global__ void k_sims(const float* __restrict__ mem, const float* __restrict__ q,
                       const float* __restrict__ scal, float* __restrict__ sims) {
  const int lane = threadIdx.x & 31;
  const int wave = threadIdx.x >> 5;
  const int base = (blockIdx.x * 8 + wave) * 16;   // 512 blocks * 8 waves * 16 rows = 65536
  const int half = lane >> 4;
  const int m    = lane & 15;
  const float* rp = mem + (size_t)(base + m) * H + 4 * half;
  const float qn  = scal[1];
  const float s0  = (lane == 0)  ? 1.0f : 0.0f;    // selector: B column-0, K sub-pair 0
  const float s16 = (lane == 16) ? 1.0f : 0.0f;    // selector: B column-0, K sub-pair 1
  float ssq = 0.0f;
  v8f acc = {0.f, 0.f, 0.f, 0.f, 0.f, 0.f, 0.f, 0.f};

#if defined(USE_WMMA_F32X4)
#pragma unroll 2
  for (int k = 0; k < H; k += 8) {
    if ((k & 127) == 0) __builtin_prefetch(rp + k + 256);
    const v4f d = *(const v4f*)(rp + k);           // half0: cols k..k+3 ; half1: k+4..k+7
    ssq += d[0] * d[0] + d[1] * d[1] + d[2] * d[2] + d[3] * d[3];
    const float sx = __shfl_xor(d[0], 16, 32);     // cross-half exchange
    const float sy = __shfl_xor(d[1], 16, 32);
    const float sz = __shfl_xor(d[2], 16, 32);
    const float sw = __shfl_xor(d[3], 16, 32);
    v2f a1, a2;
    a1.x = half ? sz   : d[0];   a1.y = half ? sw   : d[1];   // A for K = k..k+3
    a2.x = half ? d[2] : sx;     a2.y = half ? d[3] : sy;     // A for K = k+4..k+7
    const float q0 = q[k],     q1 = q[k + 1], q2 = q[k + 2], q3 = q[k + 3];
    const float q4 = q[k + 4], q5 = q[k + 5], q6 = q[k + 6], q7 = q[k + 7];
    v2f b1, b2;                                    // q in column 0 only, branchless
    b1.x = s0 * q0 + s16 * q2;   b1.y = s0 * q1 + s16 * q3;
    b2.x = s0 * q4 + s16 * q6;   b2.y = s0 * q5 + s16 * q7;
    acc = __builtin_amdgcn_wmma_f32_16x16x4_f32(false, a1, false, b1, (short)0, acc,
                                                false, false);
    acc = __builtin_amdgcn_wmma_f32_16x16x4_f32(false, a2, false, b2, (short)0, acc,
                                                false, false);
  }
#else
  // host-pass placeholder (never executed on device builds)
  for (int k = 0; k < H; ++k) { acc[0] += rp[k] * q[k]; ssq += rp[k] * rp[k]; }
#endif

  ssq += __shfl_xor(ssq, 16, 32);   // halves cover cols 0..3 / 4..7 mod 8 -> full row
#pragma unroll
  for (int r = 0; r < 16; ++r) {
    const float s2 = __shfl(ssq, r, 32);
    const float dd = (r < 8) ? __shfl(acc[r], 0, 32) : __shfl(acc[r - 8], 16, 32);
    if (lane == 0) sims[base + r] = dd / (sqrtf(s2) * qn + EPSF);
  }
}

// ---- K6: LCA over 65536 slots; one workgroup, v in registers ----
__global__ __launch_bounds__(1024) void k_lca(const float* __restrict__ sims,
                                              const float* __restrict__ scal,
                                              float* __restrict__ wts) {
  __shared__ float red[32];
  const int t = threadIdx.x;
  const float win = scal[0];
  float v[64];
#pragma unroll
  for (int e = 0; e < 64; ++e) v[e] = 0.0f;
  for (int cyc = 0; cyc < NCYC; ++cyc) {
    float p = 0.0f;
#pragma unroll
    for (int e = 0; e < 64; ++e) p += fmaxf(v[e], 0.0f);
    const float S = blockRed(p, red);
#pragma unroll
    for (int e = 0; e < 64; ++e) {
      const int j = t + e * 1024;
      const float inp = win * sims[j];
      const float fv  = fmaxf(v[e], 0.0f);
      const float dv  = DT_F * (inp - CMPT * (S - fv));   // leak = 0
      v[e] = fmaxf(v[e] + dv, 0.0f);
    }
  }
#pragma unroll
  for (int e = 0; e < 64; ++e) wts[t + e * 1024] = v[e];
}

// ---- K7: partial m_t = sum_r wts[r]*mem[r]  fused with mem_new copy ----
// Rows walked in REVERSE so the tail of mem (still resident in the 192MB L2
// from k_sims) is re-hit instead of LRU-thrashed. Streaming NT stores.
// (mem_new in d_out is only 4B aligned -> scalar NT stores.)
__global__ void k_msum_copy(const float* __restrict__ mem, const float* __restrict__ wts,
                            float* __restrict__ mtpart, float* __restrict__ memnew) {
  const int t = threadIdx.x;                       // 256 threads = 1 row of float4
  const int b = blockIdx.x;                        // 256 blocks
  const int rbase = (int)(gridDim.x - 1 - b) * 256;
  const v4f* m4 = (const v4f*)mem;
  v4f acc = {0.f, 0.f, 0.f, 0.f};
  for (int r = 255; r >= 0; --r) {
    const int row = rbase + r;
    const float wv = wts[row];
    const v4f d = __builtin_nontemporal_load(&m4[(size_t)row * 256 + t]);
    acc += d * wv;
    float* o = memnew + (size_t)row * H + t * 4;
    __builtin_nontemporal_store(d[0], &o[0]);
    __builtin_nontemporal_store(d[1], &o[1]);
    __builtin_nontemporal_store(d[2], &o[2]);
    __builtin_nontemporal_store(d[3], &o[3]);
  }
  float* mp = mtpart + (size_t)b * 1024 + t * 4;
  mp[0] = acc[0]; mp[1] = acc[1]; mp[2] = acc[2]; mp[3] = acc[3];
}

// ---- K8: m_t, cm_t, h_t2, mem_new[write_ptr] row ----
__global__ __launch_bounds__(1024) void k_final_state(const float* __restrict__ mtpart,
                                                      const float* __restrict__ c_t,
                                                      const float* __restrict__ pre,
                                                      const int* __restrict__ wp,
                                                      float* __restrict__ out,
                                                      float* __restrict__ ht2) {
  const int i = threadIdx.x;
  float m = 0.0f;
  for (int b = 0; b < 256; ++b) m += mtpart[(size_t)b * 1024 + i];
  const float cm = c_t[i] + m;
  const float o  = sigm(pre[H + i]);
  const float h2 = o * tanhf(cm);
  out[OUT_MT + i] = m;
  out[OUT_CM + i] = cm;
  out[OUT_H2 + i] = h2;
  ht2[i] = h2;
  out[OUT_MEM + (size_t)wp[0] * H + i] = cm;
}

// ---- K10: actor/critic heads + softmax ----
__global__ void k_heads(const float* __restrict__ dec2, const float* __restrict__ Wa,
                        const float* __restrict__ ba, const float* __restrict__ Wc,
                        const float* __restrict__ bc, float* __restrict__ out) {
  __shared__ float red[32];
  __shared__ float res[5];
  const int t = threadIdx.x;
  for (int row = 0; row < 5; ++row) {
    const float* w = (row < 4) ? (Wa + (size_t)row * H) : Wc;
    float p = 0.0f;
    for (int j = t; j < H; j += 256) p += w[j] * dec2[j];
    p = blockRed(p, red);
    if (t == 0) res[row] = p + ((row < 4) ? ba[row] : bc[0]);
  }
  __syncthreads();
  if (t == 0) {
    const float mx = fmaxf(fmaxf(res[0], res[1]), fmaxf(res[2], res[3]));
    const float e0 = expf(res[0] - mx), e1 = expf(res[1] - mx);
    const float e2 = expf(res[2] - mx), e3 = expf(res[3] - mx);
    const float s = e0 + e1 + e2 + e3;
    out[OUT_PI + 0] = e0 / s; out[OUT_PI + 1] = e1 / s;
    out[OUT_PI + 2] = e2 / s; out[OUT_PI + 3] = e3 / s;
    out[OUT_VAL] = res[4];
  }
}

extern "C" void kernel_launch(void* const* d_in, const int* in_sizes, int n_in,
                              void* d_out, int out_size, void* d_ws, size_t ws_size,
                              hipStream_t stream) {
  const float* x      = (const float*)d_in[0];
  const float* hprev  = (const float*)d_in[1];
  const float* cprev  = (const float*)d_in[2];
  const float* mem    = (const float*)d_in[3];
  const float* Wi2h   = (const float*)d_in[4];
  const float* bi2h   = (const float*)d_in[5];
  const float* Wh2h   = (const float*)d_in[6];
  const float* bh2h   = (const float*)d_in[7];
  const float* Wih    = (const float*)d_in[8];
  const float* bih    = (const float*)d_in[9];
  const float* Wactor = (const float*)d_in[10];
  const float* bactor = (const float*)d_in[11];
  const float* Wcrit  = (const float*)d_in[12];
  const float* bcrit  = (const float*)d_in[13];
  const float* Whpc   = (const float*)d_in[14];
  const float* bhpc   = (const float*)d_in[15];
  const int*   wp     = (const int*)d_in[16];

  float* out = (float*)d_out;
  float* ws  = (float*)d_ws;

  float* pre    = ws + WS_PREACT;
  float* ct     = ws + WS_CT;
  float* ht     = ws + WS_HT;
  float* dec    = ws + WS_DEC;
  float* scal   = ws + WS_SCAL;
  float* sims   = ws + WS_SIMS;
  float* wts    = ws + WS_WTS;
  float* mtpart = ws + WS_MTPART;
  float* dec2   = ws + WS_DEC2;
  float* ht2    = ws + WS_HT2;

  k_preact<<<(G4 + 7) / 8, 256, 0, stream>>>(Wi2h, bi2h, Wh2h, bh2h, x, hprev, pre);
  k_gates<<<1, 1024, 0, stream>>>(pre, cprev, ct, ht, scal);
  k_matvec_relu<<<H / 8, 256, 0, stream>>>(Wih, bih, ht, dec, H, H);
  k_phi<<<1, 256, 0, stream>>>(ct, dec, Whpc, bhpc, scal);
  k_sims<<<DLEN / 128, 256, 0, stream>>>(mem, ct, scal, sims);       // 512 blocks
  k_lca<<<1, 1024, 0, stream>>>(sims, scal, wts);
  k_msum_copy<<<DLEN / 256, 256, 0, stream>>>(mem, wts, mtpart, out + OUT_MEM);
  k_final_state<<<1, 1024, 0, stream>>>(mtpart, ct, pre, wp, out, ht2);
  k_matvec_relu<<<H / 8, 256, 0, stream>>>(Wih, bih, ht2, dec2, H, H);
  k_heads<<<1, 256, 0, stream>>>(dec2, Wactor, bactor, Wcrit, bcrit, out);
}